// GATGraphClassifier_4947802325330
// MI455X (gfx1250) — compile-verified
//
#include <hip/hip_runtime.h>

#define NNODES   100000
#define NGRAPHS  128
#define FDIM     128   // H*C
#define CDIM     64
#define HHEADS   2
#define OUTDIM   10
#define NEG_SLOPE 0.2f

typedef __attribute__((ext_vector_type(2))) float v2f;
typedef __attribute__((ext_vector_type(8))) float v8f;

// ---------------------------------------------------------------------------
// GEMM via V_WMMA_F32_16X16X4_F32.  C[M,128] = A[M,K] * B[K,128], row-major.
// Block = 256 threads (8 waves) covering one 16-row strip of A:
//   - A tile (16 x K) staged once into LDS, coalesced (read from HBM exactly once)
//   - wave w computes n-tile w (columns 16w..16w+15); B stays L0/L2-resident
// Fragment layouts (ISA 7.12.2):
//   A 16x4 f32: VGPR0 = K0|K2 across lane halves, VGPR1 = K1|K3
//   B 4x16:     row-striped across lanes, rows split across lane halves
//   C/D 16x16:  VGPR v -> M=v (lanes 0-15) / M=v+8 (lanes 16-31)
// LDS row padded +4 DWORDs -> per-lane bank stride 4, conflict-free ds_load_b64.
// ---------------------------------------------------------------------------
__global__ __launch_bounds__(256)
void gemm_f32_wmma(const float* __restrict__ A, const float* __restrict__ B,
                   float* __restrict__ Cmat, int K) {
  __shared__ float At[16][FDIM + 4];
  const int tid  = threadIdx.x;
  const int wave = tid >> 5;
  const int lane = tid & 31;
  const int half = lane >> 4;      // 0 or 1
  const int l16  = lane & 15;
  const int m0   = blockIdx.x * 16;

  // cooperative, coalesced stage of the 16 x K A tile (K/16 floats per thread)
  {
    const int r  = tid >> 4;                 // 0..15
    const int c0 = (tid & 15) * (K >> 4);    // K/16 consecutive floats
    const float* ap = A + (long)(m0 + r) * K + c0;
    for (int i = 0; i < (K >> 4); i += 4)
      *(float4*)(&At[r][c0 + i]) = *(const float4*)(ap + i);
  }
  __syncthreads();

  const int nb = wave * 16;                  // this wave's n-tile base
  v8f c = {0.f, 0.f, 0.f, 0.f, 0.f, 0.f, 0.f, 0.f};
  #pragma unroll 4
  for (int k0 = 0; k0 < K; k0 += 4) {
    const int kk = k0 + 2 * half;
    v2f a = *(const v2f*)(&At[l16][kk]);     // ds_load_b64, conflict-free
    v2f b;
    b.x = B[(long)kk       * FDIM + nb + l16];
    b.y = B[(long)(kk + 1) * FDIM + nb + l16];
    c = __builtin_amdgcn_wmma_f32_16x16x4_f32(
            /*neg_a=*/false, a, /*neg_b=*/false, b,
            /*c_mod=*/(short)0, c, /*reuse_a=*/false, /*reuse_b=*/false);
  }
  const int crow = m0 + half * 8;
  float* cp = Cmat + (long)crow * FDIM + nb + l16;
  #pragma unroll
  for (int v = 0; v < 8; ++v) cp[(long)v * FDIM] = c[v];
}

// a_src[n,h] = dot(xw[n,h,:], att_src[h,:]);  a_dst likewise. 1 thread / (n,h).
// xw is L2-resident (51 MB of 192 MB L2), so this pass is cheap.
__global__ void att_kernel(const float* __restrict__ xw,
                           const float* __restrict__ att_src,
                           const float* __restrict__ att_dst,
                           float* __restrict__ a_src, float* __restrict__ a_dst) {
  int t = blockIdx.x * blockDim.x + threadIdx.x;
  if (t >= NNODES * HHEADS) return;
  const int n = t >> 1, h = t & 1;
  const float4* xp = (const float4*)(xw + (long)n * FDIM + h * CDIM);
  const float4* sp = (const float4*)(att_src + h * CDIM);
  const float4* dp = (const float4*)(att_dst + h * CDIM);
  float as = 0.f, ad = 0.f;
  #pragma unroll
  for (int i = 0; i < CDIM / 4; ++i) {
    float4 x = xp[i], s = sp[i], d = dp[i];
    as += x.x * s.x + x.y * s.y + x.z * s.z + x.w * s.w;
    ad += x.x * d.x + x.y * d.y + x.z * d.z + x.w * d.w;
  }
  a_src[t] = as; a_dst[t] = ad;
}

// zero acc[N*128], set m = -inf, s = 0 (d_ws is poisoned 0xAA by harness)
__global__ void init_layer(float* __restrict__ m, float* __restrict__ s,
                           float* __restrict__ acc) {
  int t = blockIdx.x * blockDim.x + threadIdx.x;
  if (t < NNODES * FDIM) acc[t] = 0.f;
  if (t < NNODES * HHEADS) { m[t] = -__builtin_inff(); s[t] = 0.f; }
}

__device__ __forceinline__ void edge_sd(int e, int E,
                                        const int* __restrict__ e0,
                                        const int* __restrict__ e1,
                                        int& src, int& dst) {
  if (e < E) { src = e0[e]; dst = e1[e]; }
  else       { src = dst = e - E; }          // self-loops appended
}

// order-preserving float atomic max (monotone int/uint bit trick)
__device__ __forceinline__ void atomicMaxF(float* addr, float val) {
  if (val >= 0.f) atomicMax((int*)addr, __float_as_int(val));
  else            atomicMin((unsigned int*)addr, __float_as_uint(val));
}

// pass 1: logits = leaky_relu(a_src[src] + a_dst[dst]); segment max over dst
__global__ void edge_logits_max(const int* __restrict__ e0, const int* __restrict__ e1,
                                int E, int EE,
                                const float* __restrict__ a_src,
                                const float* __restrict__ a_dst,
                                float* __restrict__ elog, float* __restrict__ m) {
  int e = blockIdx.x * blockDim.x + threadIdx.x;
  if (e >= EE) return;
  int src, dst; edge_sd(e, E, e0, e1, src, dst);
  #pragma unroll
  for (int h = 0; h < HHEADS; ++h) {
    float l = a_src[src * HHEADS + h] + a_dst[dst * HHEADS + h];
    l = (l > 0.f) ? l : NEG_SLOPE * l;
    elog[(long)e * HHEADS + h] = l;
    atomicMaxF(&m[dst * HHEADS + h], l);
  }
}

// pass 2: e = exp(l - m[dst]); segment sum over dst; overwrite elog with e
__global__ void edge_exp_sum(const int* __restrict__ e0, const int* __restrict__ e1,
                             int E, int EE,
                             float* __restrict__ elog,
                             const float* __restrict__ m, float* __restrict__ s) {
  int e = blockIdx.x * blockDim.x + threadIdx.x;
  if (e >= EE) return;
  int src, dst; edge_sd(e, E, e0, e1, src, dst);
  (void)src;
  #pragma unroll
  for (int h = 0; h < HHEADS; ++h) {
    float ex = __expf(elog[(long)e * HHEADS + h] - m[dst * HHEADS + h]);
    elog[(long)e * HHEADS + h] = ex;
    atomicAdd(&s[dst * HHEADS + h], ex);
  }
}

// pass 3: acc[dst] += alpha * xw[src]. One wave32 per edge; lane covers 4 of
// the 128 features (lanes 0-15 -> head 0, lanes 16-31 -> head 1).
__global__ __launch_bounds__(256)
void edge_scatter(const int* __restrict__ e0, const int* __restrict__ e1,
                  int E, int EE,
                  const float* __restrict__ elog, const float* __restrict__ s,
                  const float* __restrict__ xw, float* __restrict__ acc) {
  int gt = blockIdx.x * blockDim.x + threadIdx.x;
  int e = gt >> 5;
  int lane = gt & 31;
  if (e >= EE) return;
  int src, dst; edge_sd(e, E, e0, e1, src, dst);
  const int h = lane >> 4;
  float alpha = elog[(long)e * HHEADS + h] /
                (s[dst * HHEADS + h] + 1e-16f);
  const int j0 = lane * 4;
  float4 x = *(const float4*)(xw + (long)src * FDIM + j0);
  float* op = acc + (long)dst * FDIM + j0;
  atomicAdd(op + 0, alpha * x.x);
  atomicAdd(op + 1, alpha * x.y);
  atomicAdd(op + 2, alpha * x.z);
  atomicAdd(op + 3, alpha * x.w);
}

// mean over heads + bias + relu  -> h[n, 64]
__global__ void finalize(const float* __restrict__ acc, const float* __restrict__ bias,
                         float* __restrict__ hout) {
  int t = blockIdx.x * blockDim.x + threadIdx.x;
  if (t >= NNODES * CDIM) return;
  const int n = t >> 6, c = t & 63;
  float v = 0.5f * (acc[(long)n * FDIM + c] + acc[(long)n * FDIM + CDIM + c]) + bias[c];
  hout[t] = (v > 0.f) ? v : 0.f;
}

__global__ void zero_kernel(float* __restrict__ p, int n) {
  int t = blockIdx.x * blockDim.x + threadIdx.x;
  if (t < n) p[t] = 0.f;
}

// global_add_pool via atomics
__global__ void pool_kernel(const float* __restrict__ h, const int* __restrict__ batch,
                            float* __restrict__ g) {
  int t = blockIdx.x * blockDim.x + threadIdx.x;
  if (t >= NNODES * CDIM) return;
  const int n = t >> 6, c = t & 63;
  atomicAdd(&g[(long)batch[n] * CDIM + c], h[t]);
}

// FC (64->10) + log_softmax; one thread per graph.
__global__ __launch_bounds__(128)
void fc_lsm(const float* __restrict__ g, const float* __restrict__ Wfc,
            const float* __restrict__ bfc, float* __restrict__ out) {
  int gr = threadIdx.x;               // blockDim = 128 = NGRAPHS
  float l[OUTDIM];
  #pragma unroll
  for (int o = 0; o < OUTDIM; ++o) l[o] = bfc[o];
  for (int k = 0; k < CDIM; ++k) {
    float gv = g[(long)gr * CDIM + k];
    #pragma unroll
    for (int o = 0; o < OUTDIM; ++o) l[o] += gv * Wfc[k * OUTDIM + o];
  }
  float mx = l[0];
  #pragma unroll
  for (int o = 1; o < OUTDIM; ++o) mx = fmaxf(mx, l[o]);
  float ssum = 0.f;
  #pragma unroll
  for (int o = 0; o < OUTDIM; ++o) ssum += __expf(l[o] - mx);
  float lse = mx + __logf(ssum);
  #pragma unroll
  for (int o = 0; o < OUTDIM; ++o) out[(long)gr * OUTDIM + o] = l[o] - lse;
}

// ---------------------------------------------------------------------------
extern "C" void kernel_launch(void* const* d_in, const int* in_sizes, int n_in,
                              void* d_out, int out_size, void* d_ws, size_t ws_size,
                              hipStream_t stream) {
  const float* x        = (const float*)d_in[0];
  const float* W1       = (const float*)d_in[1];
  const float* att_src1 = (const float*)d_in[2];
  const float* att_dst1 = (const float*)d_in[3];
  const float* b1       = (const float*)d_in[4];
  const float* W2       = (const float*)d_in[5];
  const float* att_src2 = (const float*)d_in[6];
  const float* att_dst2 = (const float*)d_in[7];
  const float* b2       = (const float*)d_in[8];
  const float* Wfc      = (const float*)d_in[9];
  const float* bfc      = (const float*)d_in[10];
  const int*   ei       = (const int*)d_in[11];   // [2, E]
  const int*   batch    = (const int*)d_in[12];

  const int E  = in_sizes[11] / 2;
  const int* e0 = ei;
  const int* e1 = ei + E;
  const int EE = E + NNODES;                      // + self-loops

  // workspace carve-out (floats)
  float* p     = (float*)d_ws;
  float* xw    = p; p += (size_t)NNODES * FDIM;   // 12.8M
  float* acc   = p; p += (size_t)NNODES * FDIM;   // 12.8M
  float* h1    = p; p += (size_t)NNODES * CDIM;   // 6.4M
  float* a_src = p; p += (size_t)NNODES * HHEADS;
  float* a_dst = p; p += (size_t)NNODES * HHEADS;
  float* mbuf  = p; p += (size_t)NNODES * HHEADS;
  float* sbuf  = p; p += (size_t)NNODES * HHEADS;
  float* g     = p; p += (size_t)NGRAPHS * CDIM;
  float* elog  = p; p += (size_t)EE * HHEADS;     // 3.4M

  const int gemmBlocks = NNODES / 16;             // 6250 strips
  const int B256 = 256;
  const int gNF   = (NNODES * FDIM + B256 - 1) / B256;
  const int gNH   = (NNODES * HHEADS + B256 - 1) / B256;
  const int gNC   = (NNODES * CDIM + B256 - 1) / B256;
  const int gEE   = (EE + B256 - 1) / B256;
  const int gEE32 = (int)(((long)EE * 32 + B256 - 1) / B256);

  // ---- layer 1 ----
  gemm_f32_wmma<<<gemmBlocks, 256, 0, stream>>>(x, W1, xw, FDIM);
  att_kernel<<<gNH, B256, 0, stream>>>(xw, att_src1, att_dst1, a_src, a_dst);
  init_layer<<<gNF, B256, 0, stream>>>(mbuf, sbuf, acc);
  edge_logits_max<<<gEE, B256, 0, stream>>>(e0, e1, E, EE, a_src, a_dst, elog, mbuf);
  edge_exp_sum<<<gEE, B256, 0, stream>>>(e0, e1, E, EE, elog, mbuf, sbuf);
  edge_scatter<<<gEE32, B256, 0, stream>>>(e0, e1, E, EE, elog, sbuf, xw, acc);
  finalize<<<gNC, B256, 0, stream>>>(acc, b1, h1);

  // ---- layer 2 (K = 64) ----
  gemm_f32_wmma<<<gemmBlocks, 256, 0, stream>>>(h1, W2, xw, CDIM);
  att_kernel<<<gNH, B256, 0, stream>>>(xw, att_src2, att_dst2, a_src, a_dst);
  init_layer<<<gNF, B256, 0, stream>>>(mbuf, sbuf, acc);
  edge_logits_max<<<gEE, B256, 0, stream>>>(e0, e1, E, EE, a_src, a_dst, elog, mbuf);
  edge_exp_sum<<<gEE, B256, 0, stream>>>(e0, e1, E, EE, elog, mbuf, sbuf);
  edge_scatter<<<gEE32, B256, 0, stream>>>(e0, e1, E, EE, elog, sbuf, xw, acc);
  finalize<<<gNC, B256, 0, stream>>>(acc, b2, h1);   // overwrite h1 (no longer needed)

  // ---- pool + FC + log_softmax ----
  zero_kernel<<<(NGRAPHS * CDIM + B256 - 1) / B256, B256, 0, stream>>>(g, NGRAPHS * CDIM);
  pool_kernel<<<gNC, B256, 0, stream>>>(h1, batch, g);
  fc_lsm<<<1, NGRAPHS, 0, stream>>>(g, Wfc, bfc, (float*)d_out);
}